// GEPSLinear_64098091925941
// MI455X (gfx1250) — compile-verified
//
#include <hip/hip_runtime.h>

#define B_BATCH 128
#define IN_F    1024
#define OUT_F   1024
#define FACTOR  1.0f

#define KCHUNK  64                    // K rows staged per LDS buffer
#define NTILE   32                    // output columns per block
#define NCHUNKS (IN_F / KCHUNK)       // 16

typedef float v2f __attribute__((ext_vector_type(2)));
typedef float v8f __attribute__((ext_vector_type(8)));

// gfx1250 async global->LDS copy, b32 variant (per-lane 4B, scatter-capable
// on the LDS side so we can store K-pair-interleaved). ASYNCcnt-tracked.
#if __has_builtin(__builtin_amdgcn_global_load_async_to_lds_b32)
  #define HAVE_ASYNC 1
  #define ASYNC_COPY_B32(gp, lp)                                               \
      __builtin_amdgcn_global_load_async_to_lds_b32(                           \
          (__attribute__((address_space(1))) int*)(gp),                        \
          (__attribute__((address_space(3))) int*)(lp), 0, 0)
  #define ASYNC_WAIT() asm volatile("s_wait_asynccnt 0x0" ::: "memory")
#else
  #define HAVE_ASYNC 0
#endif

// ---------------------------------------------------------------------------
// Kernel 1: per-batch rank-2 coefficients.
//   u[r]  = sum_i input[b,i] * A[i,r]
//   v[p]  = sum_r u[r] * codes[b,r,p]
//   ws4[b] = { FACTOR*v0, FACTOR*v1, FACTOR*codes[b,0,0], FACTOR*codes[b,1,1] }
// ---------------------------------------------------------------------------
__global__ __launch_bounds__(256) void geps_coeff_kernel(
    const float* __restrict__ input,
    const float* __restrict__ codes,
    const float* __restrict__ A,
    float4* __restrict__ ws4)
{
    int wave = (int)((blockIdx.x * blockDim.x + threadIdx.x) >> 5);
    int lane = (int)(threadIdx.x & 31);
    if (wave >= B_BATCH) return;

    const float* in_row = input + (size_t)wave * IN_F;
    float u0 = 0.f, u1 = 0.f;
    for (int i = lane; i < IN_F; i += 32) {
        float  x = in_row[i];
        float2 a = ((const float2*)A)[i];
        u0 += x * a.x;
        u1 += x * a.y;
    }
    for (int off = 16; off > 0; off >>= 1) {
        u0 += __shfl_down(u0, off, 32);
        u1 += __shfl_down(u1, off, 32);
    }
    if (lane == 0) {
        float c00 = codes[wave * 4 + 0];
        float c01 = codes[wave * 4 + 1];
        float c10 = codes[wave * 4 + 2];
        float c11 = codes[wave * 4 + 3];
        float v0 = u0 * c00 + u1 * c10;
        float v1 = u0 * c01 + u1 * c11;
        ws4[wave] = make_float4(FACTOR * v0, FACTOR * v1, FACTOR * c00, FACTOR * c11);
    }
}

// ---------------------------------------------------------------------------
// Kernel 2: out = input @ weight via V_WMMA_F32_16X16X4_F32.
// Block = one 32-column N-group (8 waves share it); wave = one 16-row M-tile.
// Weight tile staged once per block through double-buffered LDS chunks via
// async b32 copies in a K-pair-INTERLEAVED layout:
//     lds[(k>>1)*64 + n*2 + (k&1)] = weight[kc+k][n_base+n]
// so a B fragment {w[k][n], w[k+1][n]} is a contiguous 8B -> one ds_load_b64
// directly into the WMMA operand register pair (no repack movs).
// Fragment layout per CDNA5 ISA 7.12.2:
//   A 16x4 f32 : lanes 0-15 -> M=lane, VGPR{0,1}=K{0,1}; lanes 16-31 -> K{2,3}
//   B  4x16 f32: lanes 0-15 -> N=lane, VGPR{0,1}=K{0,1}; lanes 16-31 -> K{2,3}
//   C 16x16 f32: VGPR r -> M = r + 8*(lane>>4), N = lane & 15
// ---------------------------------------------------------------------------
__global__ __launch_bounds__(256) void geps_gemm_kernel(
    const float*  __restrict__ input,
    const float*  __restrict__ weight,
    const float*  __restrict__ Bmat,
    const float*  __restrict__ bias,
    const float*  __restrict__ bias_ctx,
    const float4* __restrict__ ws4,
    float*        __restrict__ out)
{
    __shared__ float smem[2][KCHUNK * NTILE];   // 2 x 8 KB

    const int n_base = (int)blockIdx.x * NTILE;      // 32 blocks
    const int m_base = (int)(threadIdx.x >> 5) * 16; // 8 waves -> 128 rows

    const int lane = (int)(threadIdx.x & 31);
    const int half = lane >> 4;        // 0: K=0,1   1: K=2,3
    const int l15  = lane & 15;

    v8f c0 = {};
    v8f c1 = {};

    const float* a_ptr = input + (size_t)(m_base + l15) * IN_F + 2 * half;

    // Cooperative stage of one 64x32 chunk (2048 floats): 8 b32 per thread,
    // global side coalesced (each wave reads 128 contiguous bytes), LDS side
    // K-pair-interleaved.
    auto issue_chunk = [&](int kc, int buf) {
        const int t = (int)threadIdx.x;
        #pragma unroll
        for (int j = 0; j < 8; ++j) {
            const int e = t + j * 256;     // 0..2047
            const int k = e >> 5;          // 0..63
            const int n = e & 31;          // 0..31
            const float* g = weight + (size_t)(kc + k) * OUT_F + n_base + n;
            float*       l = &smem[buf][((k >> 1) * NTILE + n) * 2 + (k & 1)];
#if HAVE_ASYNC
            ASYNC_COPY_B32(g, l);
#else
            *l = *g;
#endif
        }
    };

    issue_chunk(0, 0);
#if HAVE_ASYNC
    ASYNC_WAIT();
#endif
    __syncthreads();

    for (int c = 0; c < NCHUNKS; ++c) {
        // Prefetch next chunk into the other buffer (overlaps with WMMAs below).
        if (c + 1 < NCHUNKS) issue_chunk((c + 1) * KCHUNK, (c + 1) & 1);

        const float* bs    = &smem[c & 1][0];
        const int    kbase = c * KCHUNK;

        #pragma unroll
        for (int k2 = 0; k2 < KCHUNK; k2 += 4) {
            // A fragment: 2 consecutive K values per lane (8B-aligned global)
            v2f a = *(const v2f*)(a_ptr + kbase + k2);

            // B fragments: K-pair p = (k2 + 2*half)/2, contiguous 8B in LDS
            const int pb = ((k2 + 2 * half) >> 1) * NTILE * 2;
            v2f b0 = *(const v2f*)&bs[pb + l15 * 2];          // cols l15
            v2f b1 = *(const v2f*)&bs[pb + (l15 + 16) * 2];   // cols l15+16

            c0 = __builtin_amdgcn_wmma_f32_16x16x4_f32(
                    false, a, false, b0, (short)0, c0, false, false);
            c1 = __builtin_amdgcn_wmma_f32_16x16x4_f32(
                    false, a, false, b1, (short)0, c1, false, false);
        }

#if HAVE_ASYNC
        ASYNC_WAIT();         // own async copies into next buffer done
#endif
        __syncthreads();      // all waves' copies visible; buffer reuse safe
    }

    // ---- fused epilogue -----------------------------------------------------
    const int ncol0 = n_base + l15;
    const int ncol1 = ncol0 + 16;

    const float bi0  = bias[ncol0],              bi1  = bias[ncol1];
    const float B00  = Bmat[ncol0],              B01  = Bmat[ncol1];
    const float B10  = Bmat[OUT_F + ncol0],      B11  = Bmat[OUT_F + ncol1];
    const float bc00 = bias_ctx[ncol0],          bc01 = bias_ctx[ncol1];
    const float bc10 = bias_ctx[OUT_F + ncol0],  bc11 = bias_ctx[OUT_F + ncol1];

    #pragma unroll
    for (int r = 0; r < 8; ++r) {
        const int row = m_base + r + 8 * half;
        const float4 s = ws4[row];   // {F*v0, F*v1, F*d0, F*d1}
        out[(size_t)row * OUT_F + ncol0] =
            c0[r] + bi0 + s.x * B00 + s.y * B10 + s.z * bc00 + s.w * bc10;
        out[(size_t)row * OUT_F + ncol1] =
            c1[r] + bi1 + s.x * B01 + s.y * B11 + s.z * bc01 + s.w * bc11;
    }
}

// ---------------------------------------------------------------------------
extern "C" void kernel_launch(void* const* d_in, const int* in_sizes, int n_in,
                              void* d_out, int out_size, void* d_ws, size_t ws_size,
                              hipStream_t stream)
{
    const float* input    = (const float*)d_in[0];  // [128,1024]
    const float* codes    = (const float*)d_in[1];  // [128,2,2]
    const float* weight   = (const float*)d_in[2];  // [1024,1024]
    const float* A        = (const float*)d_in[3];  // [1024,2]
    const float* Bmat     = (const float*)d_in[4];  // [2,1024]
    const float* bias     = (const float*)d_in[5];  // [1024]
    const float* bias_ctx = (const float*)d_in[6];  // [2,1024]
    float*  out = (float*)d_out;                    // [128,1024]
    float4* ws4 = (float4*)d_ws;                    // 128 * float4 = 2 KB

    geps_coeff_kernel<<<16, 256, 0, stream>>>(input, codes, A, ws4);
    geps_gemm_kernel<<<32, 256, 0, stream>>>(input, weight, Bmat, bias,
                                             bias_ctx, ws4, out);
}